// InstanceMetricLoss_87101936763364
// MI455X (gfx1250) — compile-verified
//
#include <hip/hip_runtime.h>
#include <math.h>

// CDNA5 (gfx1250) fused InstanceMetricLoss:
//   D = X X^T / 128 computed in 16x16 tiles via V_WMMA_F32_16X16X4_F32 (exact fp32),
//   fused exp(1+D) row-sum reduction (never materializes the 8192x8192 Gram matrix),
//   per-pair 2x2 diagonal block recomputed exactly, deterministic reductions.

typedef float v2f __attribute__((ext_vector_type(2)));
typedef float v8f __attribute__((ext_vector_type(8)));

#define DM          128          // feature dim (K)
#define COLBLK      64           // columns staged in LDS per iteration
#define LDS_STRIDE  132          // 128 dwords + 4 pad: bank-conflict-free, 16B-aligned rows
#define NSLICE      4            // column slices (grid.y) -> deterministic partial sums
#define INV_COS     (1.0f/128.0f)

__global__ __launch_bounds__(256)
void gram_rowsum_kernel(const float* __restrict__ X, float* __restrict__ S,
                        int Btot, int blksPerSlice)
{
    __shared__ float tile[COLBLK * LDS_STRIDE];   // 64 x 128 column tile of X (padded)

    const int tid  = threadIdx.x;
    const int wave = tid >> 5;        // 8 waves / workgroup
    const int lane = tid & 31;
    const int lr   = lane & 15;       // 0..15  : row (A) / col (B) within tile
    const int lh   = lane >> 4;       // 0/1    : selects K-pair {0,1} vs {2,3}

    const int rowbase = blockIdx.x * 128 + wave * 16;   // this wave's 16 rows
    const int row     = rowbase + lr;

    // ---- Preload A fragments for full K=128 (32 x v2f = 64 VGPRs), kept for whole sweep.
    // Lane layout of 32-bit 16x4 A tile: M = lane%16, K = {2*(lane/16), +1} + 4*t.
    v2f areg[32];
    {
        const float* arow = X + (size_t)row * DM + 2 * lh;
        #pragma unroll
        for (int t = 0; t < 32; ++t)
            areg[t] = *(const v2f*)(arow + 4 * t);
    }

    float acc[8];
    #pragma unroll
    for (int v = 0; v < 8; ++v) acc[v] = 0.0f;

    const int blk0 = blockIdx.y * blksPerSlice;
    for (int b = 0; b < blksPerSlice; ++b) {
        const int cbase = (blk0 + b) * COLBLK;

        // ---- Stage 64x128 column tile into LDS, coalesced float4 (8 per thread).
        const float4* src = (const float4*)(X + (size_t)cbase * DM);
        __syncthreads();
        #pragma unroll
        for (int q = 0; q < (COLBLK * DM / 4) / 256; ++q) {
            int idx = tid + q * 256;          // 0..2047
            int r   = idx >> 5;               // 32 float4 per row
            int c4  = idx & 31;
            float4 dv = src[r * 32 + c4];
            *(float4*)(&tile[r * LDS_STRIDE + c4 * 4]) = dv;
        }
        // Prefetch next column tile while we compute on this one.
        if (b + 1 < blksPerSlice)
            __builtin_prefetch(X + (size_t)(cbase + COLBLK) * DM + tid * 8, 0, 1);
        __syncthreads();

        // ---- 4 column sub-tiles of 16: D-tile = sum_k A*B via 32 f32 WMMAs, then fuse exp.
        #pragma unroll 1
        for (int sub = 0; sub < COLBLK / 16; ++sub) {
            v8f c = (v8f)0.0f;
            // B 4x16 tile layout mirrors A: N = lane%16, same K-pair -> contiguous float2.
            const float* bp = &tile[(sub * 16 + lr) * LDS_STRIDE + 2 * lh];
            #pragma unroll
            for (int t = 0; t < 32; ++t) {
                v2f bfrag = *(const v2f*)(bp + 4 * t);
                c = __builtin_amdgcn_wmma_f32_16x16x4_f32(
                        false, areg[t], false, bfrag, (short)0, c, false, false);
            }
            // C layout: c[v] = D[rowbase + v + 8*lh][cbase + sub*16 + lr]
            #pragma unroll
            for (int v = 0; v < 8; ++v)
                acc[v] += __expf(1.0f + c[v] * INV_COS);
        }
    }

    // ---- Reduce across the 16 column-lanes of each half-wave (stays within 16-group).
    #pragma unroll
    for (int m = 1; m < 16; m <<= 1) {
        #pragma unroll
        for (int v = 0; v < 8; ++v)
            acc[v] += __shfl_xor(acc[v], m, 32);
    }
    // lane 0 -> rows rowbase+0..7, lane 16 -> rows rowbase+8..15 (partial: this col slice)
    if (lr == 0) {
        float* out = S + (size_t)blockIdx.y * Btot + rowbase + 8 * lh;
        #pragma unroll
        for (int v = 0; v < 8; ++v) out[v] = acc[v];
    }
}

__global__ __launch_bounds__(256)
void pair_kernel(const float* __restrict__ X, const float* __restrict__ S,
                 float* __restrict__ T, int Btot)
{
    const int p = blockIdx.x * blockDim.x + threadIdx.x;
    const int P = Btot >> 1;
    if (p >= P) return;
    const int i = 2 * p, j = 2 * p + 1;

    const float* xi = X + (size_t)i * DM;
    const float* xj = X + (size_t)j * DM;
    float sii = 0.f, sij = 0.f, sjj = 0.f;
    #pragma unroll 8
    for (int k = 0; k < DM; ++k) {
        float a = xi[k], b = xj[k];
        sii = fmaf(a, a, sii);
        sij = fmaf(a, b, sij);
        sjj = fmaf(b, b, sjj);
    }
    const float Dii = sii * INV_COS, Dij = sij * INV_COS, Djj = sjj * INV_COS;

    // Combine the NSLICE partial row sums in fixed order (deterministic).
    float Si = 0.f, Sj = 0.f;
    #pragma unroll
    for (int s = 0; s < NSLICE; ++s) {
        Si += S[(size_t)s * Btot + i];
        Sj += S[(size_t)s * Btot + j];
    }
    // Exclude the 2x2 diagonal block (D[i,j] == D[j,i] by symmetry).
    const float dissim = (Si + Sj)
        - (__expf(1.0f + Dii) + __expf(1.0f + Djj) + 2.0f * __expf(1.0f + Dij));
    const float J = __logf(1e-8f + dissim) - Dij;
    const float h = fmaxf(J, 0.0f);
    T[p] = h * h;
}

__global__ __launch_bounds__(256)
void reduce_kernel(const float* __restrict__ T, float* __restrict__ out,
                   int P, float scale)
{
    __shared__ float red[256];
    const int tid = threadIdx.x;
    float s = 0.0f;
    for (int q = tid; q < P; q += 256) s += T[q];   // fixed-order strided partials
    red[tid] = s;
    __syncthreads();
    for (int off = 128; off > 0; off >>= 1) {
        if (tid < off) red[tid] += red[tid + off];
        __syncthreads();
    }
    if (tid == 0) out[0] = red[0] * scale;
}

extern "C" void kernel_launch(void* const* d_in, const int* in_sizes, int n_in,
                              void* d_out, int out_size, void* d_ws, size_t ws_size,
                              hipStream_t stream) {
    const float* X = (const float*)d_in[0];
    const int Btot = in_sizes[0] / DM;          // 8192
    const int P    = Btot >> 1;                 // 4096 pairs

    float* S = (float*)d_ws;                    // NSLICE x Btot partial row sums
    float* T = S + (size_t)NSLICE * Btot;       // P per-pair loss terms

    const int colBlocks     = Btot / COLBLK;    // 128
    const int blksPerSlice  = colBlocks / NSLICE;

    dim3 grid(Btot / 128, NSLICE);
    gram_rowsum_kernel<<<grid, 256, 0, stream>>>(X, S, Btot, blksPerSlice);
    pair_kernel<<<(P + 255) / 256, 256, 0, stream>>>(X, S, T, Btot);
    reduce_kernel<<<1, 256, 0, stream>>>(T, (float*)d_out, P, 1.0f / (2.0f * Btot));
}